// ActorNetwork_mid_concat_28724741276371
// MI455X (gfx1250) — compile-verified
//
#include <hip/hip_runtime.h>

typedef __attribute__((ext_vector_type(2))) float v2f;
typedef __attribute__((ext_vector_type(8))) float v8f;

// One workgroup = 8 waves (wave32). Wave w owns h rows [16w, 16w+16).
__launch_bounds__(256)
__global__ void abr_policy_rollout_kernel(
    const float* __restrict__ inp,     // 8x8
    const float* __restrict__ conv_w,  // 128x1x4
    const float* __restrict__ conv_b,  // 128
    const float* __restrict__ w0, const float* __restrict__ b0,   // 128
    const float* __restrict__ w1, const float* __restrict__ b1,   // 128
    const float* __restrict__ w2, const float* __restrict__ b2,   // 128
    const float* __restrict__ w3, const float* __restrict__ b3,   // 128x2048, 128
    const float* __restrict__ w4, const float* __restrict__ b4,   // 6x128, 6
    float* __restrict__ out)           // 6
{
    __shared__ float st[6][8];
    __shared__ float bw[8];
    __shared__ float vcs[6];
    __shared__ __align__(16) float feat[2048];
    __shared__ float hraw[128];
    __shared__ float hact[128];
    __shared__ float logits[6];

    const int tid  = threadIdx.x;
    const int lane = tid & 31;
    const int wv   = tid >> 5;

    // ---- init state / bw / vcs from inp ----
    if (tid < 48)                 st[tid >> 3][tid & 7] = inp[tid];
    if (tid >= 48 && tid < 56)    bw[tid - 48] = inp[tid];
    if (tid >= 64 && tid < 70)    vcs[tid - 64] = inp[56] * (float)(1 << (tid - 64));
    __syncthreads();

    for (int step = 0; step < 7; ++step) {
        // ================= features (feat[2048] in LDS) =================
        {
            const int c = tid & 127;
            const float cw0 = conv_w[c * 4 + 0];
            const float cw1 = conv_w[c * 4 + 1];
            const float cw2 = conv_w[c * 4 + 2];
            const float cw3 = conv_w[c * 4 + 3];
            const float cb  = conv_b[c];
            if (tid < 128) {                       // waves 0..3 (uniform)
                feat[c] = fmaxf(st[0][7] * w0[c] + b0[c], 0.f);          // s0
                #pragma unroll
                for (int j = 0; j < 5; ++j) {                            // s2
                    float v = st[2][j]*cw0 + st[2][j+1]*cw1 +
                              st[2][j+2]*cw2 + st[2][j+3]*cw3 + cb;
                    feat[256 + c * 5 + j] = fmaxf(v, 0.f);
                }
                #pragma unroll
                for (int j = 0; j < 3; ++j) {                            // s4
                    float v = st[4][j]*cw0 + st[4][j+1]*cw1 +
                              st[4][j+2]*cw2 + st[4][j+3]*cw3 + cb;
                    feat[1536 + c * 3 + j] = fmaxf(v, 0.f);
                }
            } else {                               // waves 4..7 (uniform)
                feat[128 + c] = fmaxf(st[1][7] * w1[c] + b1[c], 0.f);    // s1
                #pragma unroll
                for (int j = 0; j < 5; ++j) {                            // s3
                    float v = st[3][j]*cw0 + st[3][j+1]*cw1 +
                              st[3][j+2]*cw2 + st[3][j+3]*cw3 + cb;
                    feat[896 + c * 5 + j] = fmaxf(v, 0.f);
                }
                feat[1920 + c] = st[4][7] * w2[c] + b2[c];               // s5 (no relu)
            }
        }
        __syncthreads();

        // ========== h = w3 @ feat via V_WMMA_F32_16X16X4_F32 ==========
        // A: w3 16x4 tile (lanes 0-15 -> K0/K1, lanes 16-31 -> K2/K3)
        // B: feat chunk broadcast across all 16 columns (same koff layout)
        {
            const int row  = wv * 16 + (lane & 15);
            const int koff = (lane >> 4) << 1;            // 0 or 2
            const float* wrow = w3 + (size_t)row * 2048 + koff;
            const float* fko  = &feat[koff];

            v8f acc0 = {0,0,0,0,0,0,0,0};
            v8f acc1 = {0,0,0,0,0,0,0,0};
            v8f acc2 = {0,0,0,0,0,0,0,0};
            v8f acc3 = {0,0,0,0,0,0,0,0};

            for (int k = 0; k < 2048; k += 16) {
                v2f a0 = *(const v2f*)(wrow + k + 0);
                v2f a1 = *(const v2f*)(wrow + k + 4);
                v2f a2 = *(const v2f*)(wrow + k + 8);
                v2f a3 = *(const v2f*)(wrow + k + 12);
                v2f f0 = *(const v2f*)(fko + k + 0);
                v2f f1 = *(const v2f*)(fko + k + 4);
                v2f f2 = *(const v2f*)(fko + k + 8);
                v2f f3 = *(const v2f*)(fko + k + 12);
                acc0 = __builtin_amdgcn_wmma_f32_16x16x4_f32(
                        false, a0, false, f0, (short)0, acc0, false, false);
                acc1 = __builtin_amdgcn_wmma_f32_16x16x4_f32(
                        false, a1, false, f1, (short)0, acc1, false, false);
                acc2 = __builtin_amdgcn_wmma_f32_16x16x4_f32(
                        false, a2, false, f2, (short)0, acc2, false, false);
                acc3 = __builtin_amdgcn_wmma_f32_16x16x4_f32(
                        false, a3, false, f3, (short)0, acc3, false, false);
            }
            v8f acc = acc0 + acc1 + acc2 + acc3;

            // D layout: lanes 0-15 hold M=v, lanes 16-31 hold M=v+8 (all N equal)
            if ((lane & 15) == 0) {
                const int mb = wv * 16 + (lane >> 4) * 8;
                #pragma unroll
                for (int v = 0; v < 8; ++v) hraw[mb + v] = acc[v];
            }
        }
        __syncthreads();

        if (tid < 128) hact[tid] = fmaxf(hraw[tid] + b3[tid], 0.f);
        __syncthreads();

        // ================= head: logits = w4 @ h + b4 =================
        if (tid < 6) {
            float s = b4[tid];
            const float* wr = w4 + tid * 128;
            #pragma unroll 8
            for (int j = 0; j < 128; ++j) s += wr[j] * hact[j];
            logits[tid] = s;
            if (step == 6) out[tid] = s;
        }
        __syncthreads();

        // ================= argmax + state transition =================
        if (step < 6) {
            if (tid == 0) {
                int a = 0; float best = logits[0];
                #pragma unroll
                for (int j = 1; j < 6; ++j)
                    if (logits[j] > best) { best = logits[j]; a = j; }
                // roll left (circular) along axis 1
                #pragma unroll
                for (int r = 0; r < 6; ++r) {
                    float t0 = st[r][0];
                    #pragma unroll
                    for (int c2 = 0; c2 < 7; ++c2) st[r][c2] = st[r][c2 + 1];
                    st[r][7] = t0;
                }
                const float VBR[6] = {300.f, 750.f, 1200.f, 1850.f, 2850.f, 4300.f};
                float delay = vcs[a] / bw[step] - 30000.0f;
                st[0][7] = VBR[a] / 4300.0f;
                st[1][7] = 30000.0f / 1000.0f / 10.0f;
                st[2][7] = vcs[a] / delay / 1000.0f;
                st[3][7] = delay / 1000.0f / 10.0f;
                #pragma unroll
                for (int j = 0; j < 6; ++j) st[4][j] = vcs[j] / 1000000.0f;
                st[5][7] = (7.0f - (float)step) / 8.0f;
            }
            __syncthreads();
        }
    }
}

extern "C" void kernel_launch(void* const* d_in, const int* in_sizes, int n_in,
                              void* d_out, int out_size, void* d_ws, size_t ws_size,
                              hipStream_t stream) {
    (void)in_sizes; (void)n_in; (void)out_size; (void)d_ws; (void)ws_size;
    abr_policy_rollout_kernel<<<1, 256, 0, stream>>>(
        (const float*)d_in[0],  // inp
        (const float*)d_in[1],  // conv_w
        (const float*)d_in[2],  // conv_b
        (const float*)d_in[3], (const float*)d_in[4],   // w0 b0
        (const float*)d_in[5], (const float*)d_in[6],   // w1 b1
        (const float*)d_in[7], (const float*)d_in[8],   // w2 b2
        (const float*)d_in[9], (const float*)d_in[10],  // w3 b3
        (const float*)d_in[11], (const float*)d_in[12], // w4 b4
        (float*)d_out);
}